// Hierarchy_molemap_logits_47253230191392
// MI455X (gfx1250) — compile-verified
//
#include <hip/hip_runtime.h>
#include <math.h>

// CDNA5 wave32 ext-vector fragment types for V_WMMA_F32_16X16X4_F32
typedef float v2f __attribute__((ext_vector_type(2)));
typedef float v8f __attribute__((ext_vector_type(8)));

#define ROWS   128   // rows per block (N = 524288 is divisible by 128)
#define NT     128   // threads per block = 4 waves (wave32)
#define L      65    // leaves per row
#define NNODE  76    // node_probs per row
#define NEXTRA 11    // p_lvl2(8) + p_top(2) + root(1)

#if __has_builtin(__builtin_amdgcn_global_load_async_to_lds_b128)
#define HAVE_ASYNC_LDS 1
// Builtin expects: (vector_size(16) int addrspace(1)*, ... lds ptr ..., imm, imm)
typedef int v4i_vs __attribute__((vector_size(16)));
typedef __attribute__((address_space(1))) v4i_vs* gb128_t;
typedef __attribute__((address_space(3))) v4i_vs* lb128_t;
#else
#define HAVE_ASYNC_LDS 0
#endif

__device__ __forceinline__ void wait_asynccnt0() {
#if __has_builtin(__builtin_amdgcn_s_wait_asynccnt)
  __builtin_amdgcn_s_wait_asynccnt(0);
#else
  asm volatile("s_wait_asynccnt 0x0" ::: "memory");
#endif
}

// group index of a leaf: boundaries 7,30,48,53,58,59,62 (branchless)
__device__ __forceinline__ int gidx_of(int col) {
  return (col >= 7) + (col >= 30) + (col >= 48) + (col >= 53) +
         (col >= 58) + (col >= 59) + (col >= 62);
}

__global__ __launch_bounds__(NT) void molemap_fwd(
    const float* __restrict__ x,
    float* __restrict__ out_leaf,    // [N,65] leaf_path_probs
    float* __restrict__ out_node) {  // [N,76] node_probs
  __shared__ float lds_x[ROWS * L];         // x tile; overwritten with leaf_probs
  __shared__ float lds_sum[ROWS * 8];       // WMMA group sums
  __shared__ float lds_fac[ROWS * 8];       // p_lvl2[g] * p_top[tidx(g)]
  __shared__ float lds_extra[ROWS * NEXTRA];

  const int t = threadIdx.x;
  const size_t row0 = (size_t)blockIdx.x * ROWS;

  // ---- Stage 1: stage the block's x tile into LDS (16B granules) ----
  // Preferred path: CDNA5 async memory->LDS DMA (no VGPR round trip,
  // tracked by ASYNCcnt). Fallback: plain float4 copy through VGPRs.
  {
    const float4* __restrict__ src = (const float4*)(x + row0 * L);
    const int total4 = ROWS * L / 4;  // 2080 float4 per block
#if HAVE_ASYNC_LDS
    float4* lp = (float4*)lds_x;
    for (int i = t; i < total4; i += NT) {
      __builtin_amdgcn_global_load_async_to_lds_b128(
          (gb128_t)(src + i), (lb128_t)(lp + i), 0, 0);
    }
    wait_asynccnt0();   // async LDS writes are NOT covered by the barrier
#else
    float4* dst = (float4*)lds_x;
    for (int i = t; i < total4; i += NT) dst[i] = src[i];
#endif
  }
  __syncthreads();

  // ---- Stage 2: per-wave WMMA f32 16x16x4 computing 8 group sums / row ----
  // sums[16 rows, 8 groups] = A(16x65 x-tile, K padded to 68) x B(65x8 0/1 mask)
  {
    const int lane  = t & 31;
    const int m     = lane & 15;            // A row within tile / B,D column
    const int khalf = (lane >> 4) * 2;      // K sub-offset (ISA A/B layout)
    const int wave  = t >> 5;
    for (int tile = 0; tile < 2; ++tile) {
      const int rbase = wave * 32 + tile * 16;
      const float* xrow = &lds_x[(rbase + m) * L];
      v8f c = {};
      #pragma unroll
      for (int s = 0; s < 17; ++s) {
        const int k0 = s * 4 + khalf;
        v2f a, b;
        a.x = (k0     < L) ? xrow[k0]     : 0.0f;
        a.y = (k0 + 1 < L) ? xrow[k0 + 1] : 0.0f;
        // B[k][n] = 1 iff leaf k belongs to group n (n = lane&15, valid n<8)
        b.x = (m < 8 && k0     < L && gidx_of(k0)     == m) ? 1.0f : 0.0f;
        b.y = (m < 8 && k0 + 1 < L && gidx_of(k0 + 1) == m) ? 1.0f : 0.0f;
        // 8 args: (neg_a, A, neg_b, B, c_mod, C, reuse_a, reuse_b)
        c = __builtin_amdgcn_wmma_f32_16x16x4_f32(false, a, false, b,
                                                  (short)0, c, false, false);
      }
      // D layout: lane l, VGPR r -> M = rbase + r + 8*(l>=16), N = l&15
      if (m < 8) {
        const int radd = (lane >> 4) * 8;
        #pragma unroll
        for (int r = 0; r < 8; ++r)
          lds_sum[(rbase + radd + r) * 8 + m] = c[r];
      }
    }
  }
  __syncthreads();

  // ---- Stage 3: per-thread row math (thread t owns row t) ----
  {
    const int row = t;
    float* xr = &lds_x[row * L];
    float s[8];
    #pragma unroll
    for (int g = 0; g < 8; ++g) s[g] = lds_sum[row * 8 + g];

    const float inv_sz[8] = {1.f/7.f, 1.f/23.f, 1.f/18.f, 1.f/5.f,
                             1.f/5.f, 1.f,      1.f/3.f,  1.f/3.f};
    float mean[8];
    #pragma unroll
    for (int g = 0; g < 8; ++g) mean[g] = s[g] * inv_sz[g];

    // level-1 (top) softmax over [mean(x[0:53]), mean(x[53:65])]
    const float mt0 = (s[0] + s[1] + s[2] + s[3]) * (1.0f / 53.0f);
    const float mt1 = (s[4] + s[5] + s[6] + s[7]) * (1.0f / 12.0f);
    const float tmax = fmaxf(mt0, mt1);
    const float e0 = __expf(mt0 - tmax), e1 = __expf(mt1 - tmax);
    const float rinv = 1.0f / (e0 + e1);
    const float pt0 = e0 * rinv, pt1 = e1 * rinv;

    // level-2 softmaxes over means[0:4] and means[4:8]
    float p2[8];
    {
      const float m0 = fmaxf(fmaxf(mean[0], mean[1]), fmaxf(mean[2], mean[3]));
      const float m1 = fmaxf(fmaxf(mean[4], mean[5]), fmaxf(mean[6], mean[7]));
      float s0 = 0.f, s1 = 0.f;
      #pragma unroll
      for (int g = 0; g < 4; ++g) { p2[g] = __expf(mean[g] - m0); s0 += p2[g]; }
      #pragma unroll
      for (int g = 4; g < 8; ++g) { p2[g] = __expf(mean[g] - m1); s1 += p2[g]; }
      const float r0 = 1.f / s0, r1 = 1.f / s1;
      #pragma unroll
      for (int g = 0; g < 4; ++g) p2[g] *= r0;
      #pragma unroll
      for (int g = 4; g < 8; ++g) p2[g] *= r1;
    }

    #pragma unroll
    for (int g = 0; g < 8; ++g) {
      lds_fac[row * 8 + g] = p2[g] * ((g < 4) ? pt0 : pt1);
      lds_extra[row * NEXTRA + g] = p2[g];
    }
    lds_extra[row * NEXTRA + 8]  = pt0;
    lds_extra[row * NEXTRA + 9]  = pt1;
    lds_extra[row * NEXTRA + 10] = 1.0f;

    // leaf-level softmax within each sibling group, in place in LDS
    const int GA[9] = {0, 7, 30, 48, 53, 58, 59, 62, 65};
    #pragma unroll
    for (int g = 0; g < 8; ++g) {
      const int a = GA[g], b = GA[g + 1];
      float mx = -INFINITY;
      for (int j = a; j < b; ++j) mx = fmaxf(mx, xr[j]);
      float sum = 0.f;
      for (int j = a; j < b; ++j) {
        const float e = __expf(xr[j] - mx);
        xr[j] = e;
        sum += e;
      }
      const float rs = 1.f / sum;
      for (int j = a; j < b; ++j) xr[j] *= rs;
    }
  }
  __syncthreads();

  // ---- Stage 4: coalesced float4 stores of both output regions ----
  {
    // leaf_path_probs = leaf_probs * fac[gidx(leaf)]
    float4* dstL = (float4*)(out_leaf + row0 * L);
    const int totL4 = ROWS * L / 4;  // 2080
    for (int i = t; i < totL4; i += NT) {
      float vv[4];
      #pragma unroll
      for (int c = 0; c < 4; ++c) {
        const int idx = i * 4 + c;
        const int row = idx / L;
        const int col = idx - row * L;
        vv[c] = lds_x[idx] * lds_fac[row * 8 + gidx_of(col)];
      }
      dstL[i] = make_float4(vv[0], vv[1], vv[2], vv[3]);
    }
    // node_probs = [leaf_probs(65) | p_lvl2(8) | p_top(2) | 1]
    float4* dstN = (float4*)(out_node + row0 * NNODE);
    const int totN4 = ROWS * NNODE / 4;  // 2432
    for (int i = t; i < totN4; i += NT) {
      float vv[4];
      #pragma unroll
      for (int c = 0; c < 4; ++c) {
        const int idx = i * 4 + c;
        const int row = idx / NNODE;
        const int col = idx - row * NNODE;
        vv[c] = (col < L) ? lds_x[row * L + col]
                          : lds_extra[row * NEXTRA + (col - L)];
      }
      dstN[i] = make_float4(vv[0], vv[1], vv[2], vv[3]);
    }
  }
}

extern "C" void kernel_launch(void* const* d_in, const int* in_sizes, int n_in,
                              void* d_out, int out_size, void* d_ws, size_t ws_size,
                              hipStream_t stream) {
  (void)n_in; (void)out_size; (void)d_ws; (void)ws_size;
  const float* x = (const float*)d_in[0];
  const int n = in_sizes[0] / L;                    // 524288 rows
  float* out_leaf = (float*)d_out;                  // [N,65]
  float* out_node = (float*)d_out + (size_t)n * L;  // [N,76]
  const int blocks = n / ROWS;                      // 4096
  molemap_fwd<<<blocks, NT, 0, stream>>>(x, out_leaf, out_node);
}